// DetectionLayer_27023934226529
// MI455X (gfx1250) — compile-verified
//
#include <hip/hip_runtime.h>
#include <stdint.h>

// ---------------- problem constants (from reference) ----------------
#define SIDE      28
#define NUM       5
#define CLASSES   80
#define COORDS    4
#define BATCHSZ   512
#define CH        (NUM * (COORDS + 1) + CLASSES)      // 105 channels per cell
#define CELLS_IMG (SIDE * SIDE)                       // 784
#define CELLS_TOT (BATCHSZ * CELLS_IMG)               // 401408

#define LOC_ELEMS  ((size_t)BATCHSZ * CELLS_IMG * NUM * 4)        // 8,028,160
#define CLS_ELEMS  ((size_t)BATCHSZ * CELLS_IMG * NUM * CLASSES)  // 160,563,200

// ---------------- launch geometry ----------------
#define CPB   128                      // cells per block (thread-per-cell)
#define TPB   128                      // 4 wave32s
#define LDS_FLOATS   (CPB * CH)        // 13,440 floats = 53,760 B
#define V4_PER_BLOCK (LDS_FLOATS / 4)  // 3,360 16-byte packets

typedef __attribute__((ext_vector_type(4))) float v4f;

__device__ __forceinline__ float fast_sigmoid(float x) {
    // 1 / (1 + e^-x) using v_exp_f32 + v_rcp_f32 (plenty of precision here)
    return __builtin_amdgcn_rcpf(1.0f + __expf(-x));
}

__global__ __launch_bounds__(TPB) void detect_head_kernel(
    const float* __restrict__ in,      // (B, 28, 28, 105) flat
    float* __restrict__ out_loc,       // (B, 3920, 4)
    float* __restrict__ out_cls,       // (B, 3920, 80)
    float* __restrict__ out_conf)      // (B, 3920)
{
    // Single shared object -> sits at LDS allocation offset 0, so raw byte
    // offsets can be fed directly to the async-DMA instruction's VDST operand.
    __shared__ __align__(16) float lds[LDS_FLOATS];

    const int t     = threadIdx.x;
    const int blk   = blockIdx.x;
    const size_t blockFloatBase = (size_t)blk * LDS_FLOATS;

    // ---------------- Phase 1: async global -> LDS (ASYNCcnt path) ----------
    {
        const char* gbase = (const char*)(in + blockFloatBase);
        for (int f = t; f < V4_PER_BLOCK; f += TPB) {
            unsigned lds_off = (unsigned)(f * 16);
            unsigned long long gaddr =
                (unsigned long long)(uintptr_t)(gbase + (size_t)f * 16);
            // LDS byte offset in lds_off; 64-bit global source address in gaddr
            asm volatile("global_load_async_to_lds_b128 %0, %1, off"
                         :: "v"(lds_off), "v"(gaddr)
                         : "memory");
        }
        asm volatile("s_wait_asynccnt 0" ::: "memory");
    }
    __syncthreads();

    // ---------------- Phase 2: per-cell math ----------------
    const int cell = blk * CPB + t;          // global cell id, < 401408 exactly
    const int s    = cell % CELLS_IMG;       // cell within image
    const float gx = (float)(s % SIDE);
    const float gy = (float)(s / SIDE);

    float* my = lds + t * CH;                // stride 105 (odd) -> bank-conflict-free

    // Stable softmax over 80 classes; keep unnormalized e^x in LDS and fold
    // the 1/sum into the per-anchor confidence scale.
    float mx = -3.402823466e38f;
    #pragma unroll 8
    for (int k = 0; k < CLASSES; ++k)
        mx = fmaxf(mx, my[NUM * (COORDS + 1) + k]);

    float sum = 0.0f;
    #pragma unroll 8
    for (int k = 0; k < CLASSES; ++k) {
        float e = __expf(my[NUM * (COORDS + 1) + k] - mx);
        my[NUM * (COORDS + 1) + k] = e;      // own slot only: no races
        sum += e;
    }
    const float rinv = __builtin_amdgcn_rcpf(sum);

    const size_t abase = (size_t)cell * NUM; // row index into (B*3920)

    float conf[NUM];
    #pragma unroll
    for (int n = 0; n < NUM; ++n) {
        const float tx = my[n * 5 + 0];
        const float ty = my[n * 5 + 1];
        const float tw = my[n * 5 + 2];
        const float th = my[n * 5 + 3];
        const float tc = my[n * 5 + 4];

        const float sx = fast_sigmoid(tx);
        const float sy = fast_sigmoid(ty);
        const float sw = fast_sigmoid(tw);
        const float sh = fast_sigmoid(th);
        const float c  = fast_sigmoid(tc);
        conf[n] = c;

        const float cx = (sx + gx) * (1.0f / SIDE);
        const float cy = (sy + gy) * (1.0f / SIDE);

        v4f box = { cx - 0.5f * sw, cy - 0.5f * sh,
                    cx + 0.5f * sw, cy + 0.5f * sh };
        __builtin_nontemporal_store(box, (v4f*)(out_loc + (abase + n) * 4));
        __builtin_nontemporal_store(c, out_conf + abase + n);
    }

    // ---------------- Phase 3: cls = softmax * conf, 642 MB NT-streamed -----
    #pragma unroll
    for (int n = 0; n < NUM; ++n) {
        const float scale = conf[n] * rinv;
        float* dst = out_cls + (abase + n) * CLASSES;  // 320B-aligned rows
        #pragma unroll 5
        for (int k = 0; k < CLASSES; k += 4) {
            const float* p = my + NUM * (COORDS + 1) + k;
            v4f v = { p[0] * scale, p[1] * scale, p[2] * scale, p[3] * scale };
            __builtin_nontemporal_store(v, (v4f*)(dst + k));
        }
    }
}

extern "C" void kernel_launch(void* const* d_in, const int* in_sizes, int n_in,
                              void* d_out, int out_size, void* d_ws, size_t ws_size,
                              hipStream_t stream) {
    const float* in = (const float*)d_in[0];
    float* out      = (float*)d_out;

    float* out_loc  = out;
    float* out_cls  = out + LOC_ELEMS;
    float* out_conf = out + LOC_ELEMS + CLS_ELEMS;

    const int grid = CELLS_TOT / CPB;   // 401408 / 128 = 3136, exact
    hipLaunchKernelGGL(detect_head_kernel, dim3(grid), dim3(TPB), 0, stream,
                       in, out_loc, out_cls, out_conf);
}